// TransformerTwinLoss_67851893342437
// MI455X (gfx1250) — compile-verified
//
#include <hip/hip_runtime.h>

// ---------------- problem constants (match reference) ----------------
#define B_DIM      512
#define T_DIM      1024
#define F_DIM      64
#define S_DIM      64
#define N_EVENTS   5
#define N_INTERV   20
#define LATENT     128
#define EPS_F      1e-8f

#define RECON_BLOCKS  1024
#define RECON_THREADS 256
#define NWAVES (RECON_BLOCKS * RECON_THREADS / 32)   // 8192 wave partials

typedef float v2f __attribute__((ext_vector_type(2)));
typedef float v8f __attribute__((ext_vector_type(8)));

// -------------------------------------------------------------------------
// Grand sum of one float per lane across the wave using V_WMMA_F32_16X16X4_F32.
// Stage 1: D1 = A(data) x ones  -> row-sums broadcast across N.
// Stage 2: accumulate ones x (pairs of D1 VGPRs) -> sums all 16 rows.
// A grand sum is layout-independent: whatever slot each value lands in,
// stage1+stage2 sum every slot exactly once. Result broadcast in all lanes.
// Requires full-wave EXEC (callers run it from a complete 32-lane wave).
// -------------------------------------------------------------------------
__device__ __forceinline__ float wave_grand_sum_wmma(float v) {
  v2f a;    a[0] = v;    a[1] = 0.0f;
  v2f ones; ones[0] = 1.0f; ones[1] = 1.0f;
  v8f d1 = {0.f, 0.f, 0.f, 0.f, 0.f, 0.f, 0.f, 0.f};
  d1 = __builtin_amdgcn_wmma_f32_16x16x4_f32(false, a, false, ones,
                                             (short)0, d1, false, false);
  v8f d2 = {0.f, 0.f, 0.f, 0.f, 0.f, 0.f, 0.f, 0.f};
#pragma unroll
  for (int i = 0; i < 4; ++i) {
    v2f b; b[0] = d1[2 * i]; b[1] = d1[2 * i + 1];
    d2 = __builtin_amdgcn_wmma_f32_16x16x4_f32(false, ones, false, b,
                                               (short)0, d2, false, false);
  }
  return d2[0];
}

// -------------------------------------------------------------------------
// Kernel 1: streaming masked-reconstruction partial sums.
// One wave per row-pair: lanes 0-15 cover row 2p, lanes 16-31 cover row 2p+1,
// each lane loads float4 from each of the 3 tensors (3x b128, fully coalesced).
// -------------------------------------------------------------------------
__global__ void __launch_bounds__(RECON_THREADS)
recon_kernel(const float* __restrict__ rec, const float* __restrict__ tgt,
             const float* __restrict__ msk, float* __restrict__ partials) {
  const int lane   = threadIdx.x & 31;
  const int wave   = (blockIdx.x * blockDim.x + threadIdx.x) >> 5;
  const int nwaves = (gridDim.x * blockDim.x) >> 5;
  const int npairs = (B_DIM * T_DIM) / 2;   // 262144 row pairs

  float acc_num = 0.0f;   // sum over rows of (rowsum diff^2) * (rowsum mask)
  float acc_den = 0.0f;   // sum over rows of (rowsum mask)

  for (int p = wave; p < npairs; p += nwaves) {
    const long long base = (long long)p * (2 * F_DIM) + lane * 4;
    const float4 r = *(const float4*)(rec + base);
    const float4 t = *(const float4*)(tgt + base);
    const float4 m = *(const float4*)(msk + base);

    const float dx = r.x - t.x, dy = r.y - t.y;
    const float dz = r.z - t.z, dw = r.w - t.w;
    float d2 = dx * dx + dy * dy + dz * dz + dw * dw;
    float ms = (m.x + m.y) + (m.z + m.w);

    // reduce within each 16-lane half (one row per half)
    d2 += __shfl_xor(d2, 1, 32);  ms += __shfl_xor(ms, 1, 32);
    d2 += __shfl_xor(d2, 2, 32);  ms += __shfl_xor(ms, 2, 32);
    d2 += __shfl_xor(d2, 4, 32);  ms += __shfl_xor(ms, 4, 32);
    d2 += __shfl_xor(d2, 8, 32);  ms += __shfl_xor(ms, 8, 32);

    if ((lane & 15) == 0) {       // lanes 0 and 16 own the two rows
      acc_num += d2 * ms;
      acc_den += ms;
    }
  }
  // combine lane 0 + lane 16
  acc_num += __shfl_xor(acc_num, 16, 32);
  acc_den += __shfl_xor(acc_den, 16, 32);
  if (lane == 0) {
    partials[2 * wave]     = acc_num;
    partials[2 * wave + 1] = acc_den;
  }
}

// -------------------------------------------------------------------------
// Kernel 2: the three small loss terms. gridDim.x == 3.
// out4: [0]=state_num [1]=state_den [2]=sum(ll) [3]=sum(1+lv-mu^2-e^lv)
// -------------------------------------------------------------------------
__global__ void __launch_bounds__(256)
small_losses_kernel(const float* __restrict__ sp, const float* __restrict__ st,
                    const float* __restrict__ sm, const float* __restrict__ hz,
                    const float* __restrict__ times, const float* __restrict__ ind,
                    const float* __restrict__ mu, const float* __restrict__ lv,
                    float* __restrict__ out4) {
  __shared__ float s0[256];
  __shared__ float s1[256];
  const int tid = threadIdx.x;

  if (blockIdx.x == 0) {
    // masked state MSE
    float num = 0.0f, den = 0.0f;
    for (int i = tid; i < B_DIM * S_DIM; i += 256) {
      const float d = sp[i] - st[i];
      const float m = sm[i];
      num += d * d * m;
      den += m;
    }
    s0[tid] = num; s1[tid] = den;
    __syncthreads();
    if (tid < 32) {
      float a = 0.0f, b = 0.0f;
#pragma unroll
      for (int j = 0; j < 8; ++j) { a += s0[tid + 32 * j]; b += s1[tid + 32 * j]; }
      const float tn = wave_grand_sum_wmma(a);
      const float td = wave_grand_sum_wmma(b);
      if (tid == 0) { out4[0] = tn; out4[1] = td; }
    }
  } else if (blockIdx.x == 1) {
    // discrete survival log-likelihood, summed over all (event, batch)
    float sll = 0.0f;
    for (int i = tid; i < N_EVENTS * B_DIM; i += 256) {
      const int e = i / B_DIM;
      const int b = i % B_DIM;
      const float tm = times[b * N_EVENTS + e];
      const float iv = ind[b * N_EVENTS + e];
      // searchsorted(bounds[1:], t, left): count of bounds[1+j] < t, clip to K-1
      int idx = 0;
#pragma unroll
      for (int j = 0; j < N_INTERV; ++j)
        idx += (0.5f * (float)(j + 1) < tm) ? 1 : 0;
      if (idx > N_INTERV - 1) idx = N_INTERV - 1;

      const float* h = hz + ((size_t)e * B_DIM + b) * N_INTERV;
      float ls = 0.0f;
      for (int j = 0; j < idx; ++j) {
        const float pj = 1.0f / (1.0f + expf(-h[j]));
        ls += logf(1.0f - pj + EPS_F);
      }
      const float pi = 1.0f / (1.0f + expf(-h[idx]));
      const float lp = logf(pi + EPS_F);
      sll += (iv > 0.5f) ? (ls + lp) : ls;
    }
    s0[tid] = sll;
    __syncthreads();
    if (tid < 32) {
      float a = 0.0f;
#pragma unroll
      for (int j = 0; j < 8; ++j) a += s0[tid + 32 * j];
      const float t = wave_grand_sum_wmma(a);
      if (tid == 0) out4[2] = t;
    }
  } else {
    // KL term: sum(1 + logvar - mu^2 - exp(logvar))
    float s = 0.0f;
    for (int i = tid; i < B_DIM * LATENT; i += 256) {
      const float m = mu[i], l = lv[i];
      s += 1.0f + l - m * m - expf(l);
    }
    s0[tid] = s;
    __syncthreads();
    if (tid < 32) {
      float a = 0.0f;
#pragma unroll
      for (int j = 0; j < 8; ++j) a += s0[tid + 32 * j];
      const float t = wave_grand_sum_wmma(a);
      if (tid == 0) out4[3] = t;
    }
  }
}

// -------------------------------------------------------------------------
// Kernel 3: reduce wave partials, combine all four terms.
// ws layout (floats): [0, 2*NWAVES) recon partials; [2*NWAVES, +4) small terms
// -------------------------------------------------------------------------
__global__ void __launch_bounds__(256)
finalize_kernel(const float* __restrict__ ws, float* __restrict__ out) {
  __shared__ float s0[256];
  __shared__ float s1[256];
  const int tid = threadIdx.x;
  float n = 0.0f, d = 0.0f;
  for (int i = tid; i < NWAVES; i += 256) {
    n += ws[2 * i];
    d += ws[2 * i + 1];
  }
  s0[tid] = n; s1[tid] = d;
  __syncthreads();
  if (tid < 32) {
    float a = 0.0f, b = 0.0f;
#pragma unroll
    for (int j = 0; j < 8; ++j) { a += s0[tid + 32 * j]; b += s1[tid + 32 * j]; }
    const float num_tot = wave_grand_sum_wmma(a);
    const float den_tot = wave_grand_sum_wmma(b);
    if (tid == 0) {
      const float* r4 = ws + 2 * NWAVES;
      const float recon = (num_tot * (1.0f / (float)F_DIM)) / (den_tot + EPS_F);
      const float state = r4[0] / (r4[1] + EPS_F);
      const float surv  = -r4[2] / (float)(N_EVENTS * B_DIM);
      const float kl    = -0.5f * r4[3] / (float)(B_DIM * LATENT);
      out[0] = recon + state + surv + 0.1f * kl;
    }
  }
}

// -------------------------------------------------------------------------
extern "C" void kernel_launch(void* const* d_in, const int* in_sizes, int n_in,
                              void* d_out, int out_size, void* d_ws, size_t ws_size,
                              hipStream_t stream) {
  (void)in_sizes; (void)n_in; (void)out_size; (void)ws_size;
  const float* rec   = (const float*)d_in[0];
  const float* tgt   = (const float*)d_in[1];
  const float* msk   = (const float*)d_in[2];
  const float* sp    = (const float*)d_in[3];
  const float* st    = (const float*)d_in[4];
  const float* sm    = (const float*)d_in[5];
  const float* hz    = (const float*)d_in[6];
  const float* times = (const float*)d_in[7];
  const float* ind   = (const float*)d_in[8];
  const float* mu    = (const float*)d_in[9];
  const float* lv    = (const float*)d_in[10];
  float* ws  = (float*)d_ws;
  float* out = (float*)d_out;

  recon_kernel<<<RECON_BLOCKS, RECON_THREADS, 0, stream>>>(rec, tgt, msk, ws);
  small_losses_kernel<<<3, 256, 0, stream>>>(sp, st, sm, hz, times, ind, mu, lv,
                                             ws + 2 * NWAVES);
  finalize_kernel<<<1, 256, 0, stream>>>(ws, out);
}